// HolographicCompressor_21990232555933
// MI455X (gfx1250) — compile-verified
//
#include <hip/hip_runtime.h>
#include <math.h>

#define D_MODEL 1024
#define STATE   128
#define BATCH   4
#define SEQ     4096
#define M_TOK   (BATCH*SEQ)          // 16384 rows
#define RMS_EPS 1.1920929e-07f

typedef __attribute__((ext_vector_type(2))) float v2f;
typedef __attribute__((ext_vector_type(8))) float v8f;

// gfx1250 async global->LDS path (ASYNCcnt), guarded so compile stays clean
// on toolchains that lack the builtins.
#if defined(__has_builtin)
#if __has_builtin(__builtin_amdgcn_global_load_async_to_lds_b32) && \
    __has_builtin(__builtin_amdgcn_s_wait_asynccnt)
#define HAVE_ASYNC_LDS 1
#endif
#endif

#if defined(HAVE_ASYNC_LDS)
typedef __attribute__((address_space(1))) int  as1_i32;
typedef __attribute__((address_space(1))) void as1_void;
typedef __attribute__((address_space(3))) int  as3_i32;
typedef __attribute__((address_space(3))) void as3_void;

// Builtin signature (per hipcc diagnostic): param0 = AS1 int*, so strip const
// and retype through void in the target address space.
__device__ __forceinline__ void async_copy_b32(const float* g, float* l) {
    __builtin_amdgcn_global_load_async_to_lds_b32(
        (as1_i32*)(as1_void*)(void*)g,
        (as3_i32*)(as3_void*)l,
        0, 0);
}
#endif

__device__ __forceinline__ float sigmoidf_(float v) {
    return 1.0f / (1.0f + __expf(-v));
}

// ---------------------------------------------------------------------------
// Stage 1: x[m,d] = emb[tokens[m], d]
// ---------------------------------------------------------------------------
__global__ void k_gather(const int* __restrict__ tok,
                         const float* __restrict__ emb,
                         float* __restrict__ x) {
    int idx = blockIdx.x * 256 + threadIdx.x;      // over M_TOK*D_MODEL (exact)
    int m = idx >> 10;
    int d = idx & 1023;
    int t = tok[m];
    x[idx] = emb[(size_t)t * D_MODEL + d];
}

// ---------------------------------------------------------------------------
// Stage 2: Bt[m,n] = sigmoid( sum_d x[m,d] * W_B[n,d] )   (WMMA f32 16x16x4)
// One wave computes one 16x16 output tile. K-loop over D_MODEL in steps of 4.
// A 16x4 f32 layout: lanes 0-15 -> K={k0,k0+1}, lanes 16-31 -> K={k0+2,k0+3}.
// ---------------------------------------------------------------------------
__global__ void __launch_bounds__(256)
k_gemm_b(const float* __restrict__ x,
         const float* __restrict__ wb,       // [STATE, D_MODEL] row-major
         float* __restrict__ bt) {           // [M_TOK, STATE]
    int wave = blockIdx.x * (blockDim.x >> 5) + (threadIdx.x >> 5);
    int lane = threadIdx.x & 31;
    int tn = wave & 7;            // STATE/16 = 8 tiles
    int tm = wave >> 3;           // M_TOK/16 = 1024 tiles
    int l16 = lane & 15;
    int kh  = lane >> 4;          // 0 or 1

    const float* arow = x  + (size_t)(tm * 16 + l16) * D_MODEL;
    const float* brow = wb + (size_t)(tn * 16 + l16) * D_MODEL;

    v8f acc = {0.f,0.f,0.f,0.f,0.f,0.f,0.f,0.f};
    for (int k0 = 0; k0 < D_MODEL; k0 += 4) {
        int kk = k0 + kh * 2;
        v2f a = *(const v2f*)(arow + kk);
        v2f b = *(const v2f*)(brow + kk);
        acc = __builtin_amdgcn_wmma_f32_16x16x4_f32(
            false, a, false, b, (short)0, acc, false, false);
    }
#pragma unroll
    for (int r = 0; r < 8; ++r) {
        int m = tm * 16 + r + kh * 8;
        int n = tn * 16 + l16;
        bt[(size_t)m * STATE + n] = sigmoidf_(acc[r]);
    }
}

// ---------------------------------------------------------------------------
// Stage 3: recurrent scan. 64 workgroups of 8 waves.
// Workgroup = (b, d-group of 256, n-chunk of 32). Wave owns 32 d, lane owns n.
// h and sigmoid(A_log) live in 64 VGPRs/lane across all 4096 steps.
// x and Bt blocks staged into LDS via async global->LDS loads when available.
// Emits per-(d-group) partial sums of h over d into P[b][dg][t][n].
// ---------------------------------------------------------------------------
__global__ void __launch_bounds__(256)
k_scan(const float* __restrict__ x,          // [M_TOK, D_MODEL]
       const float* __restrict__ bt,         // [M_TOK, STATE]
       const float* __restrict__ A_log,      // [D_MODEL, STATE]
       float* __restrict__ P) {              // [BATCH, 4, SEQ, STATE]
    __shared__ float xbuf[16][256];
    __shared__ float btbuf[16][32];
    __shared__ float hmacc[16][32];
    float* hmf = &hmacc[0][0];
    float* btf = &btbuf[0][0];

    int bid = blockIdx.x;            // 64 blocks
    int b   = bid >> 4;
    int dg  = (bid >> 2) & 3;
    int nc  = bid & 3;
    int tid  = threadIdx.x;
    int wave = tid >> 5;
    int lane = tid & 31;
    int dbase = dg * 256 + wave * 32;
    int n = nc * 32 + lane;

    float A[32], h[32];
#pragma unroll
    for (int k = 0; k < 32; ++k) {
        A[k] = sigmoidf_(A_log[(size_t)(dbase + k) * STATE + n]);
        h[k] = 0.0f;
    }

    const float* xb  = x  + (size_t)b * SEQ * D_MODEL + dg * 256;
    const float* btb = bt + (size_t)b * SEQ * STATE + nc * 32;

    for (int t0 = 0; t0 < SEQ; t0 += 16) {
        __syncthreads();                       // prev block's LDS reads done

        // ---- stage x block [16 t][256 d] and Bt block [16 t][32 n] ----
#if defined(HAVE_ASYNC_LDS)
#pragma unroll
        for (int k = 0; k < 16; ++k)
            async_copy_b32(&xb[(size_t)(t0 + k) * D_MODEL + tid],
                           &xbuf[k][tid]);
#pragma unroll
        for (int j = 0; j < 2; ++j) {
            int idx = tid * 2 + j;
            int tt = idx >> 5, ln = idx & 31;
            async_copy_b32(&btb[(size_t)(t0 + tt) * STATE + ln], &btf[idx]);
        }
#else
#pragma unroll
        for (int k = 0; k < 16; ++k)
            xbuf[k][tid] = xb[(size_t)(t0 + k) * D_MODEL + tid];
#pragma unroll
        for (int j = 0; j < 2; ++j) {
            int idx = tid * 2 + j;
            int tt = idx >> 5, ln = idx & 31;
            btf[idx] = btb[(size_t)(t0 + tt) * STATE + ln];
        }
#endif
        hmf[tid * 2]     = 0.0f;
        hmf[tid * 2 + 1] = 0.0f;
        if (t0 + 16 < SEQ)                     // gfx1250 global_prefetch_b8
            __builtin_prefetch(&xb[(size_t)(t0 + 16) * D_MODEL + tid], 0, 1);
#if defined(HAVE_ASYNC_LDS)
        __builtin_amdgcn_s_wait_asynccnt(0);
#endif
        __syncthreads();

        // ---- 16 recurrence steps from LDS ----
        for (int tt = 0; tt < 16; ++tt) {
            float btv = btbuf[tt][lane];
            const float4* xv4 = (const float4*)&xbuf[tt][wave * 32];
            float p = 0.0f;
#pragma unroll
            for (int q = 0; q < 8; ++q) {
                float4 xv = xv4[q];
                h[4*q+0] = fmaf(h[4*q+0], A[4*q+0], btv * xv.x); p += h[4*q+0];
                h[4*q+1] = fmaf(h[4*q+1], A[4*q+1], btv * xv.y); p += h[4*q+1];
                h[4*q+2] = fmaf(h[4*q+2], A[4*q+2], btv * xv.z); p += h[4*q+2];
                h[4*q+3] = fmaf(h[4*q+3], A[4*q+3], btv * xv.w); p += h[4*q+3];
            }
            atomicAdd(&hmacc[tt][lane], p);    // ds_add_f32
        }
        __syncthreads();

        // ---- write per-(d-group) partial sums ----
#pragma unroll
        for (int j = 0; j < 2; ++j) {
            int idx = tid * 2 + j;
            int tt = idx >> 5, ln = idx & 31;
            P[(size_t)((b * 4 + dg) * SEQ + t0 + tt) * STATE + nc * 32 + ln] = hmf[idx];
        }
    }
}

// ---------------------------------------------------------------------------
// Stage 4: hm[b,t,n] = (1/D) * sum_dg P[b,dg,t,n]
// ---------------------------------------------------------------------------
__global__ void k_combine(const float* __restrict__ P, float* __restrict__ hm) {
    int i = blockIdx.x * 256 + threadIdx.x;   // over BATCH*SEQ*STATE (exact)
    int b = i / (SEQ * STATE);
    int r = i - b * (SEQ * STATE);
    float s = 0.0f;
#pragma unroll
    for (int dg = 0; dg < 4; ++dg)
        s += P[(size_t)(b * 4 + dg) * (SEQ * STATE) + r];
    hm[i] = s * (1.0f / (float)D_MODEL);
}

// ---------------------------------------------------------------------------
// Stage 5: y[m,d] = sum_n hm[m,n]*W_C[d,n] + Dp[d]*x[m,d]   (WMMA f32 16x16x4)
// ---------------------------------------------------------------------------
__global__ void __launch_bounds__(256)
k_gemm_c(const float* __restrict__ hm,       // [M_TOK, STATE]
         const float* __restrict__ wc,       // [D_MODEL, STATE] row-major
         const float* __restrict__ x,
         const float* __restrict__ Dp,
         float* __restrict__ y) {            // [M_TOK, D_MODEL]
    int wave = blockIdx.x * (blockDim.x >> 5) + (threadIdx.x >> 5);
    int lane = threadIdx.x & 31;
    int td = wave & 63;           // D_MODEL/16 = 64 tiles
    int tm = wave >> 6;           // M_TOK/16 = 1024 tiles
    int l16 = lane & 15;
    int kh  = lane >> 4;

    const float* arow = hm + (size_t)(tm * 16 + l16) * STATE;
    const float* brow = wc + (size_t)(td * 16 + l16) * STATE;

    v8f acc = {0.f,0.f,0.f,0.f,0.f,0.f,0.f,0.f};
    for (int k0 = 0; k0 < STATE; k0 += 4) {
        int kk = k0 + kh * 2;
        v2f a = *(const v2f*)(arow + kk);
        v2f b = *(const v2f*)(brow + kk);
        acc = __builtin_amdgcn_wmma_f32_16x16x4_f32(
            false, a, false, b, (short)0, acc, false, false);
    }
#pragma unroll
    for (int r = 0; r < 8; ++r) {
        int m = tm * 16 + r + kh * 8;
        int d = td * 16 + l16;
        size_t o = (size_t)m * D_MODEL + d;
        y[o] = acc[r] + Dp[d] * x[o];
    }
}

// ---------------------------------------------------------------------------
// Stage 6: RMSNorm over last dim, in place on y.
// ---------------------------------------------------------------------------
__global__ void __launch_bounds__(256)
k_rmsnorm(float* __restrict__ y, const float* __restrict__ w) {
    __shared__ float red[256];
    int m = blockIdx.x, tid = threadIdx.x;
    float* row = y + (size_t)m * D_MODEL;
    float v[4];
    float s = 0.0f;
#pragma unroll
    for (int j = 0; j < 4; ++j) {
        v[j] = row[tid + 256 * j];
        s += v[j] * v[j];
    }
    red[tid] = s;
    __syncthreads();
    for (int st = 128; st > 0; st >>= 1) {
        if (tid < st) red[tid] += red[tid + st];
        __syncthreads();
    }
    float scale = rsqrtf(red[0] * (1.0f / (float)D_MODEL) + RMS_EPS);
#pragma unroll
    for (int j = 0; j < 4; ++j)
        row[tid + 256 * j] = v[j] * scale * w[tid + 256 * j];
}

// ---------------------------------------------------------------------------
extern "C" void kernel_launch(void* const* d_in, const int* in_sizes, int n_in,
                              void* d_out, int out_size, void* d_ws, size_t ws_size,
                              hipStream_t stream) {
    (void)in_sizes; (void)n_in; (void)out_size; (void)ws_size;
    const int*   tok   = (const int*)d_in[0];
    const float* emb   = (const float*)d_in[1];
    const float* A_log = (const float*)d_in[2];
    const float* W_B   = (const float*)d_in[3];
    const float* W_C   = (const float*)d_in[4];
    const float* Dp    = (const float*)d_in[5];
    const float* nw    = (const float*)d_in[6];
    float* out = (float*)d_out;

    char* ws = (char*)d_ws;
    size_t off = 0;
    float* x  = (float*)(ws + off); off += (size_t)M_TOK * D_MODEL * sizeof(float);   // 64 MiB
    float* bt = (float*)(ws + off); off += (size_t)M_TOK * STATE * sizeof(float);     //  8 MiB
    float* P  = (float*)(ws + off); off += (size_t)BATCH * 4 * SEQ * STATE * sizeof(float); // 32 MiB
    float* hm = (float*)(ws + off);                                                   //  8 MiB

    k_gather <<< (M_TOK * D_MODEL) / 256, 256, 0, stream >>> (tok, emb, x);
    k_gemm_b <<< (M_TOK / 16) * (STATE / 16) / 8, 256, 0, stream >>> (x, W_B, bt);
    k_scan   <<< 64, 256, 0, stream >>> (x, bt, A_log, P);
    k_combine<<< (BATCH * SEQ * STATE) / 256, 256, 0, stream >>> (P, hm);
    k_gemm_c <<< (M_TOK / 16) * (D_MODEL / 16) / 8, 256, 0, stream >>> (hm, W_C, x, Dp, out);
    k_rmsnorm<<< M_TOK, 256, 0, stream >>> (out, nw);
}